// GATSparse_60129542144440
// MI455X (gfx1250) — compile-verified
//
#include <hip/hip_runtime.h>
#include <hip/hip_bf16.h>

#define N_NODES 50000        // divisible by 16 -> no tile guards needed
#define E_CFG_N 800000       // divisible by 64
#define E_GKT_N 800000

typedef __attribute__((ext_vector_type(2))) float v2f;
typedef __attribute__((ext_vector_type(8))) float v8f;

// ---------------- monotone float<->uint mapping for atomicMax on floats ----
__device__ __forceinline__ unsigned f2mono(float f) {
    unsigned u = __float_as_uint(f);
    return (u & 0x80000000u) ? ~u : (u | 0x80000000u);
}
__device__ __forceinline__ float mono2f(unsigned u) {
    u = (u & 0x80000000u) ? (u & 0x7fffffffu) : ~u;
    return __uint_as_float(u);
}

// ---------------- gfx1250 async global->LDS copy (ASYNCcnt-tracked) -------
__device__ __forceinline__ void async_load_b128(unsigned lds_byte_off, const void* gptr) {
    asm volatile("global_load_async_to_lds_b128 %0, %1, off"
                 :: "v"(lds_byte_off), "v"(gptr) : "memory");
}
__device__ __forceinline__ void wait_async() {
    asm volatile("s_wait_asynccnt 0" ::: "memory");
}

// ---------------- weight repack into WMMA B-fragment order -----------------
// B fragment layout for V_WMMA_F32_16X16X4_F32 (4x16 tile, 2 VGPRs):
//   lane (hi = lane>>4, l16 = lane&15) holds pair ( W[kb][col], W[kb+1][col] )
//   with kb = 4*k4 + 2*hi, col = col0 + l16.
// Packed streams: Wm_f/Ws_f = [wave 0..7][k4 0..63][lane 0..31][2]
//                 Wa_f (=[W_a1|W_a2], 256x16) = [k4 0..63][lane][2]
//                 Wae_f (W_ae zero-padded to 128x16) = [k4 0..31][lane][2]
__global__ void repack_weights(const float* __restrict__ W_m, const float* __restrict__ W_skip,
                               const float* __restrict__ W_a1, const float* __restrict__ W_a2,
                               const float* __restrict__ W_ae,
                               float* __restrict__ Wm_f, float* __restrict__ Ws_f,
                               float* __restrict__ Wa_f, float* __restrict__ Wae_f) {
    int t = blockIdx.x * blockDim.x + threadIdx.x;
    if (t < 32768) {                       // W_m (0..16383) and W_skip (16384..32767)
        const float* src = (t < 16384) ? W_m : W_skip;
        float*       dst = (t < 16384) ? Wm_f : Ws_f;
        int p    = t & 16383;
        int w    = p >> 11;
        int k4   = (p >> 5) & 63;
        int lane = p & 31;
        int hi = lane >> 4, l16 = lane & 15;
        int kb  = 4 * k4 + 2 * hi;
        int col = 16 * w + l16;
        dst[p * 2 + 0] = src[kb * 128 + col];
        dst[p * 2 + 1] = src[(kb + 1) * 128 + col];
    } else if (t < 34816) {                // [W_a1 | W_a2] -> 256x16
        int p = t - 32768;
        int k4 = p >> 5, lane = p & 31;
        int hi = lane >> 4, l16 = lane & 15;
        int kb = 4 * k4 + 2 * hi;
        float x0, x1;
        if (l16 < 8) { x0 = W_a1[kb * 8 + l16];       x1 = W_a1[(kb + 1) * 8 + l16]; }
        else         { x0 = W_a2[kb * 8 + (l16 - 8)]; x1 = W_a2[(kb + 1) * 8 + (l16 - 8)]; }
        Wa_f[p * 2 + 0] = x0; Wa_f[p * 2 + 1] = x1;
    } else if (t < 35840) {                // W_ae padded to 128x16
        int p = t - 34816;
        int k4 = p >> 5, lane = p & 31;
        int hi = lane >> 4, l16 = lane & 15;
        int kb = 4 * k4 + 2 * hi;
        float x0 = 0.f, x1 = 0.f;
        if (l16 < 8) { x0 = W_ae[kb * 8 + l16]; x1 = W_ae[(kb + 1) * 8 + l16]; }
        Wae_f[p * 2 + 0] = x0; Wae_f[p * 2 + 1] = x1;
    }
}

// ---------------- dense per-node transforms via f32 WMMA -------------------
// One block = one 16-node tile (N_NODES % 16 == 0, no guards). 9 waves:
//   waves 0..7 : 16x16 tiles of z@W_m and z@W_skip (K=256)
//   wave  8    : 16x16 tile of z@[W_a1 | W_a2]
__global__ __launch_bounds__(288) void node_dense(
    const float* __restrict__ node_fts, const float* __restrict__ hid,
    const float* __restrict__ Wm_f, const float* __restrict__ Ws_f,
    const float* __restrict__ Wa_f,
    const float* __restrict__ b_m, const float* __restrict__ b_skip,
    const float* __restrict__ b_a1, const float* __restrict__ b_a2,
    float* __restrict__ vals, float* __restrict__ agg,
    float* __restrict__ att1, float* __restrict__ att2)
{
    __shared__ float zt[16 * 260];          // 16 rows x 256, pad 4 (bank-conflict-free)
    const int n0  = blockIdx.x * 16;
    const int tid = threadIdx.x;

    // stage z = [node_fts | hidden] via async global->LDS (16B chunks)
    for (int i = tid; i < 1024; i += 288) {
        int r = i >> 6, c = (i & 63) * 4;
        int n = n0 + r;
        unsigned lds = (unsigned)(uintptr_t)&zt[r * 260 + c];
        const float* src = (c < 128) ? (node_fts + (size_t)n * 128 + c)
                                     : (hid      + (size_t)n * 128 + (c - 128));
        async_load_b128(lds, src);
    }
    wait_async();
    __syncthreads();

    const int wave = tid >> 5;
    const int lane = tid & 31;
    const int hi   = lane >> 4;
    const int l16  = lane & 15;

    if (wave < 8) {
        const float* Bm = Wm_f + (size_t)wave * 4096;   // 64 k4-steps * 64 floats
        const float* Bs = Ws_f + (size_t)wave * 4096;
        v8f cm = {}; v8f cs = {};
        #pragma unroll 4
        for (int k4 = 0; k4 < 64; ++k4) {
            v2f a  = *(const v2f*)&zt[l16 * 260 + 4 * k4 + 2 * hi];  // ds_load_b64
            v2f bm = *(const v2f*)&Bm[(k4 * 32 + lane) * 2];         // coalesced b64
            v2f bs = *(const v2f*)&Bs[(k4 * 32 + lane) * 2];
            cm = __builtin_amdgcn_wmma_f32_16x16x4_f32(false, a, false, bm, (short)0, cm, false, false);
            cs = __builtin_amdgcn_wmma_f32_16x16x4_f32(false, a, false, bs, (short)0, cs, false, false);
        }
        const int c   = wave * 16 + l16;
        const float bmv = b_m[c];
        const float bsv = b_skip[c];
        float* vp = vals + (size_t)(n0 + 8 * hi) * 128 + c;
        float* ap = agg  + (size_t)(n0 + 8 * hi) * 128 + c;
        #pragma unroll
        for (int i = 0; i < 8; ++i) {
            vp[(size_t)i * 128] = cm[i] + bmv;
            ap[(size_t)i * 128] = cs[i] + bsv;   // skip term pre-loaded into accumulator
        }
    } else {
        v8f ca = {};
        #pragma unroll 4
        for (int k4 = 0; k4 < 64; ++k4) {
            v2f a = *(const v2f*)&zt[l16 * 260 + 4 * k4 + 2 * hi];
            v2f b = *(const v2f*)&Wa_f[(k4 * 32 + lane) * 2];
            ca = __builtin_amdgcn_wmma_f32_16x16x4_f32(false, a, false, b, (short)0, ca, false, false);
        }
        const float bav = (l16 < 8) ? b_a1[l16] : b_a2[l16 - 8];
        float* base = (l16 < 8) ? (att1 + l16) : (att2 + (l16 - 8));   // pointer select, no branch
        float* dp = base + (size_t)(n0 + 8 * hi) * 8;
        #pragma unroll
        for (int i = 0; i < 8; ++i) dp[(size_t)i * 8] = ca[i] + bav;
    }
}

// ---------------- edge-feature attention: gkt_edge_fts @ W_ae (WMMA) -------
// Block = 64 edges (E_GKT_N % 64 == 0), 4 waves, one 16-edge tile per wave.
__global__ __launch_bounds__(128) void edge_att(
    const float* __restrict__ efts, const float* __restrict__ Wae_f,
    const float* __restrict__ bae, float* __restrict__ atte)
{
    __shared__ float ft[64 * 132];
    const int e0  = blockIdx.x * 64;
    const int tid = threadIdx.x;
    for (int i = tid; i < 2048; i += 128) {          // 64 rows x 32 x 16B chunks
        int r = i >> 5, c = (i & 31) * 4;
        unsigned lds = (unsigned)(uintptr_t)&ft[r * 132 + c];
        async_load_b128(lds, efts + (size_t)(e0 + r) * 128 + c);
    }
    wait_async();
    __syncthreads();

    const int wave = tid >> 5, lane = tid & 31;
    const int hi = lane >> 4, l16 = lane & 15;
    const int rb = wave * 16;
    v8f cacc = {};
    #pragma unroll 4
    for (int k4 = 0; k4 < 32; ++k4) {
        v2f a = *(const v2f*)&ft[(rb + l16) * 132 + 4 * k4 + 2 * hi];
        v2f b = *(const v2f*)&Wae_f[(k4 * 32 + lane) * 2];
        cacc = __builtin_amdgcn_wmma_f32_16x16x4_f32(false, a, false, b, (short)0, cacc, false, false);
    }
    if (l16 < 8) {
        const float bav = bae[l16];
        float* dp = atte + (size_t)(e0 + rb + 8 * hi) * 8 + l16;
        #pragma unroll
        for (int i = 0; i < 8; ++i) dp[(size_t)i * 8] = cacc[i] + bav;
    }
}

// ---------------- sparse passes --------------------------------------------
__global__ void init_seg(unsigned* __restrict__ segmax, float* __restrict__ segsum, int n) {
    int i = blockIdx.x * blockDim.x + threadIdx.x;
    if (i < n) { segmax[i] = 0u; segsum[i] = 0.f; }
}

__global__ void edge_logits(const int* __restrict__ idx,
                            const float* __restrict__ att1, const float* __restrict__ att2,
                            const float* atte, float* ebuf,      // may alias (in-place ok)
                            unsigned* __restrict__ segmax, int nE) {
    int t = blockIdx.x * blockDim.x + threadIdx.x;
    if (t >= nE * 8) return;
    int e = t >> 3, h = t & 7;
    int row = idx[2 * e], col = idx[2 * e + 1];
    float x = att1[row * 8 + h] + att2[col * 8 + h];
    if (atte) x += atte[t];
    x = x > 0.f ? x : 0.01f * x;                  // leaky_relu, slope 0.01
    ebuf[t] = x;
    atomicMax(&segmax[row * 8 + h], f2mono(x));
}

__global__ void edge_exp(const int* __restrict__ idx, float* __restrict__ ebuf,
                         const unsigned* __restrict__ segmax, float* __restrict__ segsum, int nE) {
    int t = blockIdx.x * blockDim.x + threadIdx.x;
    if (t >= nE * 8) return;
    int e = t >> 3, h = t & 7;
    int row = idx[2 * e];
    float v = expf(ebuf[t] - mono2f(segmax[row * 8 + h]));
    ebuf[t] = v;
    atomicAdd(&segsum[row * 8 + h], v);
}

// wave-per-edge: 32 lanes x float4 = 128 columns, coalesced gather + atomics
__global__ __launch_bounds__(256) void edge_scatter(
    const int* __restrict__ idx, const float* __restrict__ ebuf,
    const float* __restrict__ segsum, const float* __restrict__ vals,
    float* __restrict__ agg, int nE) {
    int e = blockIdx.x * 8 + (threadIdx.x >> 5);
    if (e >= nE) return;
    int lane = threadIdx.x & 31;
    int row = idx[2 * e], col = idx[2 * e + 1];
    int c = lane * 4;
    int h = lane >> 2;
    float coef = ebuf[(size_t)e * 8 + h] / segsum[(size_t)row * 8 + h];
    const float4 v = *(const float4*)&vals[(size_t)col * 128 + c];
    float* dst = &agg[(size_t)row * 128 + c];
    atomicAdd(dst + 0, coef * v.x);
    atomicAdd(dst + 1, coef * v.y);
    atomicAdd(dst + 2, coef * v.z);
    atomicAdd(dst + 3, coef * v.w);
}

__global__ void relu_copy(const float* __restrict__ src, float* __restrict__ dst, int n) {
    int i = blockIdx.x * blockDim.x + threadIdx.x;
    if (i < n) dst[i] = fmaxf(src[i], 0.f);
}

// ---------------- driver ----------------------------------------------------
extern "C" void kernel_launch(void* const* d_in, const int* in_sizes, int n_in,
                              void* d_out, int out_size, void* d_ws, size_t ws_size,
                              hipStream_t stream) {
    const float* node_fts = (const float*)d_in[0];
    const float* gkt_efts = (const float*)d_in[1];
    const float* hidden   = (const float*)d_in[2];
    const int*   cfg_idx  = (const int*)d_in[3];
    const int*   gkt_idx  = (const int*)d_in[4];
    const float* W_m    = (const float*)d_in[5];
    const float* b_m    = (const float*)d_in[6];
    const float* W_skip = (const float*)d_in[7];
    const float* b_skip = (const float*)d_in[8];
    const float* W_a1   = (const float*)d_in[9];
    const float* b_a1   = (const float*)d_in[10];
    const float* W_a2   = (const float*)d_in[11];
    const float* b_a2   = (const float*)d_in[12];
    const float* W_ae   = (const float*)d_in[13];
    const float* b_ae   = (const float*)d_in[14];
    float* out = (float*)d_out;

    const int N = N_NODES, Ec = E_CFG_N, Eg = E_GKT_N;

    size_t off = 0;
    auto carve = [&](size_t bytes) {
        void* p = (char*)d_ws + off;
        off += (bytes + 255) & ~(size_t)255;
        return p;
    };
    float*    vals    = (float*)carve((size_t)N * 128 * 4);
    float*    agg     = (float*)carve((size_t)N * 128 * 4);
    float*    hidden2 = (float*)carve((size_t)N * 128 * 4);
    float*    ebuf    = (float*)carve((size_t)Eg * 8 * 4);  // logits/exp; reused as att_e
    float*    att1    = (float*)carve((size_t)N * 8 * 4);
    float*    att2    = (float*)carve((size_t)N * 8 * 4);
    unsigned* segmax  = (unsigned*)carve((size_t)N * 8 * 4);
    float*    segsum  = (float*)carve((size_t)N * 8 * 4);
    float*    Wm_f    = (float*)carve((size_t)16384 * 2 * 4);
    float*    Ws_f    = (float*)carve((size_t)16384 * 2 * 4);
    float*    Wa_f    = (float*)carve((size_t)2048 * 2 * 4);
    float*    Wae_f   = (float*)carve((size_t)1024 * 2 * 4);

    const int tilesN  = N / 16;
    const int segGrid = (N * 8 + 255) / 256;
    const int nodGrid = (N * 128 + 255) / 256;

    repack_weights<<<140, 256, 0, stream>>>(W_m, W_skip, W_a1, W_a2, W_ae,
                                            Wm_f, Ws_f, Wa_f, Wae_f);

    // ---------------- phase 1: cfg round ----------------
    node_dense<<<tilesN, 288, 0, stream>>>(node_fts, hidden, Wm_f, Ws_f, Wa_f,
                                           b_m, b_skip, b_a1, b_a2,
                                           vals, agg, att1, att2);
    init_seg<<<segGrid, 256, 0, stream>>>(segmax, segsum, N * 8);
    edge_logits<<<(Ec * 8 + 255) / 256, 256, 0, stream>>>(cfg_idx, att1, att2, nullptr, ebuf, segmax, Ec);
    edge_exp<<<(Ec * 8 + 255) / 256, 256, 0, stream>>>(cfg_idx, ebuf, segmax, segsum, Ec);
    edge_scatter<<<(Ec + 7) / 8, 256, 0, stream>>>(cfg_idx, ebuf, segsum, vals, agg, Ec);
    relu_copy<<<nodGrid, 256, 0, stream>>>(agg, hidden2, N * 128);

    // edge-feature attention for gkt (reuses ebuf as att_e storage)
    edge_att<<<Eg / 64, 128, 0, stream>>>(gkt_efts, Wae_f, b_ae, ebuf);

    // ---------------- phase 2: gkt round ----------------
    node_dense<<<tilesN, 288, 0, stream>>>(node_fts, hidden2, Wm_f, Ws_f, Wa_f,
                                           b_m, b_skip, b_a1, b_a2,
                                           vals, agg, att1, att2);
    init_seg<<<segGrid, 256, 0, stream>>>(segmax, segsum, N * 8);
    edge_logits<<<(Eg * 8 + 255) / 256, 256, 0, stream>>>(gkt_idx, att1, att2, ebuf, ebuf, segmax, Eg);
    edge_exp<<<(Eg * 8 + 255) / 256, 256, 0, stream>>>(gkt_idx, ebuf, segmax, segsum, Eg);
    edge_scatter<<<(Eg + 7) / 8, 256, 0, stream>>>(gkt_idx, ebuf, segsum, vals, agg, Eg);
    relu_copy<<<nodGrid, 256, 0, stream>>>(agg, out, N * 128);
}